// BRDFNet_21612275433506
// MI455X (gfx1250) — compile-verified
//
#include <hip/hip_runtime.h>
#include <math.h>

// ---------------------------------------------------------------------------
// Fused BRDF forward for MI455X (gfx1250, wave32).
//   conv (10x256 matvec per pixel)  -> V_WMMA_F32_16X16X4_F32, K-loop over 256
//   nonlinearities + GGX render     -> branch-free per-lane VALU
// Memory-bound: feats (536 MB) streamed once with NT loads (no L2 reuse),
// outputs (~31 MB) written with NT stores.  HBM floor ~23 us @ 23.3 TB/s.
// ---------------------------------------------------------------------------

typedef __attribute__((ext_vector_type(2))) float v2f;
typedef __attribute__((ext_vector_type(4))) float v4f;
typedef __attribute__((ext_vector_type(8))) float v8f;

#define C_DIM    256      // input channels (K)
#define PPB      65536    // pixels per batch image (256*256)
#define N_PIX    524288   // 8 * 65536 total pixels
#define BLOCK    256      // 8 waves

__device__ __forceinline__ float rcp_fast(float x)  { return __builtin_amdgcn_rcpf(x); }
__device__ __forceinline__ float sqrt_fast(float x) { return __builtin_amdgcn_sqrtf(x); }

__device__ __forceinline__ float tanh_fast(float x) {
#if __has_builtin(__builtin_amdgcn_tanhf)
    return __builtin_amdgcn_tanhf(x);          // gfx1250 V_TANH_F32
#else
    // branch-free, stable for all x: tanh(x) = sign(x)*(1-e^-2|x|)/(1+e^-2|x|)
    float ax = fabsf(x);
    float t  = __expf(-2.0f * ax);
    float r  = (1.0f - t) * rcp_fast(1.0f + t);
    return copysignf(r, x);
#endif
}

__device__ __forceinline__ float sigmoid_fast(float x) {
    return rcp_fast(1.0f + __expf(-x));
}

__global__ __launch_bounds__(BLOCK) void brdf_fused_kernel(
    const float* __restrict__ feats,     // [8][256][65536]
    const float* __restrict__ wi,        // [N_PIX][3]
    const float* __restrict__ wo,        // [N_PIX][3]
    const float* __restrict__ w_normal,  // [3][256]
    const float* __restrict__ w_albedo,  // [3][256]
    const float* __restrict__ w_rough,   // [1][256]
    const float* __restrict__ w_fresnel, // [3][256]
    float* __restrict__ imgs,            // [N_PIX][3]
    float* __restrict__ out_data)        // [N_PIX][12]
{
    // LDS weight tile: lds_w[k][m] = W[m][k], rows 10..15 zero padded.
    // Word index = k*16 + m: lanes 0-15 read k0*16+l, lanes 16-31 read
    // (k0+2)*16+l -> two disjoint 16-bank windows, conflict-free.
    __shared__ float lds_w[C_DIM][16];

    const int t = threadIdx.x;
    {
        const int c = t;                       // BLOCK == C_DIM
        lds_w[c][0] = w_normal[c];
        lds_w[c][1] = w_normal[C_DIM + c];
        lds_w[c][2] = w_normal[2 * C_DIM + c];
        lds_w[c][3] = w_albedo[c];
        lds_w[c][4] = w_albedo[C_DIM + c];
        lds_w[c][5] = w_albedo[2 * C_DIM + c];
        lds_w[c][6] = w_rough[c];
        lds_w[c][7] = w_fresnel[c];
        lds_w[c][8] = w_fresnel[C_DIM + c];
        lds_w[c][9] = w_fresnel[2 * C_DIM + c];
#pragma unroll
        for (int m = 10; m < 16; ++m) lds_w[c][m] = 0.0f;
    }
    __syncthreads();

    const int lane   = t & 31;
    const int laneLo = lane & 15;
    const int hi     = lane >> 4;                 // 0: K0/K1 half, 1: K2/K3 half

    // Each wave covers 32 consecutive pixels: tile0 = +0..15, tile1 = +16..31.
    const long gbase = (long)blockIdx.x * BLOCK + (long)(t >> 5) * 32;
    const int  batch = (int)(gbase >> 16);        // 32 | 65536 -> never straddles
    const int  p0    = (int)(gbase & 65535);

    const float* __restrict__ fb =
        feats + (size_t)batch * C_DIM * PPB + (size_t)p0 + (size_t)laneLo;

    v8f acc0 = {0.f,0.f,0.f,0.f,0.f,0.f,0.f,0.f};
    v8f acc1 = {0.f,0.f,0.f,0.f,0.f,0.f,0.f,0.f};

#pragma unroll 8
    for (int k0 = 0; k0 < C_DIM; k0 += 4) {
        const int kc = k0 + (hi << 1);            // this half-wave's K rows
        // A fragment (16x4 f32): lane M = laneLo, V0 = K[kc], V1 = K[kc+1]
        v2f a;
        a.x = lds_w[kc][laneLo];
        a.y = lds_w[kc + 1][laneLo];
        // B fragments (4x16 f32): N = laneLo, V0 = K[kc] row, V1 = K[kc+1] row
        const float* pB = fb + (size_t)kc * PPB;
        v2f b0, b1;
        b0.x = __builtin_nontemporal_load(pB);
        b0.y = __builtin_nontemporal_load(pB + PPB);
        b1.x = __builtin_nontemporal_load(pB + 16);
        b1.y = __builtin_nontemporal_load(pB + PPB + 16);
        acc0 = __builtin_amdgcn_wmma_f32_16x16x4_f32(false, a, false, b0,
                                                     (short)0, acc0, false, false);
        acc1 = __builtin_amdgcn_wmma_f32_16x16x4_f32(false, a, false, b1,
                                                     (short)0, acc1, false, false);
    }

    // ---- redistribute D fragments so each lane owns one full pixel --------
    // D layout: VGPR j, lanes 0-15 -> M=j, lanes 16-31 -> M=j+8.
    // lane L < 16  : pixel gbase+L (tile0): m0..7 own-lane, m8/m9 from L+16
    // lane L >= 16 : pixel gbase+L (tile1): m8/m9 own-lane, m0..7 from L-16
    float m0,m1,m2,m3,m4,m5,m6,m7,m8,m9;
    {
        float s0 = __shfl(acc1[0], laneLo, 32);
        float s1 = __shfl(acc1[1], laneLo, 32);
        float s2 = __shfl(acc1[2], laneLo, 32);
        float s3 = __shfl(acc1[3], laneLo, 32);
        float s4 = __shfl(acc1[4], laneLo, 32);
        float s5 = __shfl(acc1[5], laneLo, 32);
        float s6 = __shfl(acc1[6], laneLo, 32);
        float s7 = __shfl(acc1[7], laneLo, 32);
        float t0 = __shfl(acc0[0], laneLo + 16, 32);
        float t1 = __shfl(acc0[1], laneLo + 16, 32);
        m0 = hi ? s0 : acc0[0];
        m1 = hi ? s1 : acc0[1];
        m2 = hi ? s2 : acc0[2];
        m3 = hi ? s3 : acc0[3];
        m4 = hi ? s4 : acc0[4];
        m5 = hi ? s5 : acc0[5];
        m6 = hi ? s6 : acc0[6];
        m7 = hi ? s7 : acc0[7];
        m8 = hi ? acc1[0] : t0;
        m9 = hi ? acc1[1] : t1;
    }

    const long g = gbase + lane;                  // this lane's pixel

    // ---- nonlinearities (branch-free) -------------------------------------
    float nx = tanh_fast(m0), ny = tanh_fast(m1), nz = tanh_fast(m2);
    {
        float len = sqrt_fast(nx*nx + ny*ny + nz*nz);
        float inv = rcp_fast(fmaxf(len, 1e-12f));
        nx *= inv; ny *= inv; nz *= inv;
    }
    float ax = sigmoid_fast(m3), ay = sigmoid_fast(m4), az = sigmoid_fast(m5);
    float r  = fminf(fmaxf(sigmoid_fast(m6), 0.001f), 1.0f);
    float fx = sigmoid_fast(m7), fy = sigmoid_fast(m8), fz = sigmoid_fast(m9);

    // ---- GGX render -------------------------------------------------------
    const float PIf = 3.14159265358979323846f;
    float lx = __builtin_nontemporal_load(wi + g*3 + 0);
    float ly = __builtin_nontemporal_load(wi + g*3 + 1);
    float lz = __builtin_nontemporal_load(wi + g*3 + 2);
    float vx = __builtin_nontemporal_load(wo + g*3 + 0);
    float vy = __builtin_nontemporal_load(wo + g*3 + 1);
    float vz = __builtin_nontemporal_load(wo + g*3 + 2);
    {
        float il = rcp_fast(fmaxf(sqrt_fast(lx*lx + ly*ly + lz*lz), 1e-12f));
        lx *= il; ly *= il; lz *= il;
        float iv = rcp_fast(fmaxf(sqrt_fast(vx*vx + vy*vy + vz*vz), 1e-12f));
        vx *= iv; vy *= iv; vz *= iv;
    }
    float hx = lx + vx, hy = ly + vy, hz = lz + vz;
    {
        float ih = rcp_fast(fmaxf(sqrt_fast(hx*hx + hy*hy + hz*hz), 1e-12f));
        hx *= ih; hy *= ih; hz *= ih;
    }
    float NdotH = fmaxf(nx*hx + ny*hy + nz*hz, 1e-8f);
    float NdotL = fmaxf(nx*lx + ny*ly + nz*lz, 1e-8f);
    float NdotV = fmaxf(nx*vx + ny*vy + nz*vz, 1e-8f);
    float VdotH = fmaxf(vx*hx + vy*hy + vz*hz, 1e-8f);

    float alpha = r * r;
    float a2    = alpha * alpha;
    float dnm   = NdotH * NdotH * (a2 - 1.0f) + 1.0f;
    float D     = a2 * rcp_fast(PIf * dnm * dnm);
    float e     = exp2f((-5.55473f * VdotH - 6.98316f) * VdotH);
    float Fx = fx + (1.0f - fx) * e;
    float Fy = fy + (1.0f - fy) * e;
    float Fz = fz + (1.0f - fz) * e;
    float kk = alpha * 0.5f;
    float G  = rcp_fast(NdotL * (1.0f - kk) + kk) *
               rcp_fast(NdotV * (1.0f - kk) + kk);
    float sc = 0.25f * D * G;
    float inv_pi = rcp_fast(PIf);
    float ox = fmaxf((ax * (1.0f - fx) * inv_pi + sc * Fx) * NdotL, 0.0f);
    float oy = fmaxf((ay * (1.0f - fy) * inv_pi + sc * Fy) * NdotL, 0.0f);
    float oz = fmaxf((az * (1.0f - fz) * inv_pi + sc * Fz) * NdotL, 0.0f);

    // ---- stores (NT, streaming) ------------------------------------------
    __builtin_nontemporal_store(ox, imgs + g*3 + 0);
    __builtin_nontemporal_store(oy, imgs + g*3 + 1);
    __builtin_nontemporal_store(oz, imgs + g*3 + 2);

    // data: 12 floats per pixel, 48 B stride -> 16-B aligned, three b128s
    v4f* dp = (v4f*)(out_data + g*12);
    v4f d0; d0.x = nx; d0.y = ny; d0.z = nz; d0.w = ax;
    v4f d1; d1.x = ay; d1.y = az; d1.z = r;  d1.w = r;
    v4f d2; d2.x = r;  d2.y = fx; d2.z = fy; d2.w = fz;
    __builtin_nontemporal_store(d0, dp + 0);
    __builtin_nontemporal_store(d1, dp + 1);
    __builtin_nontemporal_store(d2, dp + 2);
}

extern "C" void kernel_launch(void* const* d_in, const int* in_sizes, int n_in,
                              void* d_out, int out_size, void* d_ws, size_t ws_size,
                              hipStream_t stream) {
    const float* feats     = (const float*)d_in[0];
    const float* wi        = (const float*)d_in[1];
    const float* wo        = (const float*)d_in[2];
    const float* w_normal  = (const float*)d_in[3];
    const float* w_albedo  = (const float*)d_in[4];
    const float* w_rough   = (const float*)d_in[5];
    const float* w_fresnel = (const float*)d_in[6];

    float* imgs = (float*)d_out;                       // N_PIX * 3
    float* data = (float*)d_out + (size_t)N_PIX * 3;   // N_PIX * 12

    dim3 grid(N_PIX / BLOCK);   // 2048 blocks, exact cover
    dim3 block(BLOCK);
    brdf_fused_kernel<<<grid, block, 0, stream>>>(
        feats, wi, wo, w_normal, w_albedo, w_rough, w_fresnel, imgs, data);
}